// DeformConv2d_44702019617360
// MI455X (gfx1250) — compile-verified
//
#include <hip/hip_runtime.h>
#include <hip/hip_bf16.h>

typedef __attribute__((ext_vector_type(2))) float v2f;
typedef __attribute__((ext_vector_type(8))) float v8f;

#define B_    4
#define C_    64
#define H_    128
#define W_    128
#define O_    64
#define HW_   (H_*W_)
#define KTOT  576           // 9 * 64
#define OFF_ELEMS (B_*18*H_*W_)   // 1179648
#define WT_ELEMS  (KTOT*O_)       // 36864

// ---------------------------------------------------------------------------
// Kernel 1: offset = conv3x3(x, w_off) + b_off   -> (B,18,H,W) into workspace
// ---------------------------------------------------------------------------
__global__ __launch_bounds__(256) void offset_conv_kernel(
    const float* __restrict__ x, const float* __restrict__ w_off,
    const float* __restrict__ b_off, float* __restrict__ off)
{
    int idx = blockIdx.x * 256 + threadIdx.x;       // exactly OFF_ELEMS threads
    int w  = idx & 127;
    int h  = (idx >> 7) & 127;
    int oc = (idx >> 14) % 18;
    int b  = idx / (18 << 14);
    float sum = b_off[oc];
    const float* xb = x + b * C_ * HW_;
    const float* wo = w_off + oc * C_ * 9;
    for (int c = 0; c < C_; ++c) {
        const float* xc = xb + c * HW_;
        const float* wc = wo + c * 9;
        #pragma unroll
        for (int ki = 0; ki < 3; ++ki) {
            int y = h - 1 + ki;
            if ((unsigned)y >= (unsigned)H_) continue;
            #pragma unroll
            for (int kj = 0; kj < 3; ++kj) {
                int xx = w - 1 + kj;
                if ((unsigned)xx >= (unsigned)W_) continue;
                sum = fmaf(xc[y * W_ + xx], wc[ki * 3 + kj], sum);
            }
        }
    }
    off[idx] = sum;
}

// ---------------------------------------------------------------------------
// Kernel 2: repack w_dcn[O][C][3][3] into K-pair-major layout:
//   wt[(k/2)*128 + o*2 + (k&1)] = w_dcn[o][c][kk],  k = kk*64 + c.
// A lane's B-fragment {B[k][n], B[k+1][n]} is then one contiguous,
// 8B-aligned ds_load_b64 straight into an even-aligned VGPR pair.
// ---------------------------------------------------------------------------
__global__ __launch_bounds__(256) void repack_w_kernel(
    const float* __restrict__ w_dcn, float* __restrict__ wt)
{
    int i = blockIdx.x * 256 + threadIdx.x;         // exactly WT_ELEMS threads
    int j  = i & 1;                                 // K parity
    int o  = (i >> 1) & 63;                         // output channel (column)
    int pi = i >> 7;                                // K pair index
    int k  = 2 * pi + j;
    int c  = k & 63;
    int kk = k >> 6;
    wt[i] = w_dcn[(o * C_ + c) * 9 + kk];
}

// ---------------------------------------------------------------------------
// Kernel 3: fused bilinear-gather + WMMA GEMM.
// Wave = one M-tile of 16 consecutive pixels, all 64 output channels.
// Block = 8 waves. Weights staged in LDS (576*64*4 = 147456 B) via the
// CDNA5 async Global->LDS DMA path (ASYNCcnt-tracked, no VGPR bounce).
// ---------------------------------------------------------------------------
__global__ __launch_bounds__(256) void deform_wmma_kernel(
    const float* __restrict__ x, const float* __restrict__ off,
    const float* __restrict__ wt, const float* __restrict__ b_dcn,
    float* __restrict__ out)
{
    __shared__ float wlds[KTOT * O_];

    // Async-stage weights: 36864 floats = 9216 x b128, 256 threads x 36 iters.
    // VDST operand = LDS byte offset = low 32 bits of the generic pointer to
    // the __shared__ object; the ptrtoint of &wlds[i] also escapes wlds into
    // the asm so the "memory" clobber is treated as a writer of wlds.
    {
        for (int i = threadIdx.x * 4; i < KTOT * O_; i += 256 * 4) {
            unsigned lds_off = (unsigned)(unsigned long long)&wlds[i];
            unsigned long long gaddr = (unsigned long long)(wt + i);
            asm volatile("global_load_async_to_lds_b128 %0, %1, off"
                         :: "v"(lds_off), "v"(gaddr) : "memory");
        }
        asm volatile("s_wait_asynccnt 0x0" ::: "memory");
    }
    __syncthreads();

    const int wv   = threadIdx.x >> 5;
    const int lane = threadIdx.x & 31;
    const int tile = blockIdx.x * 8 + wv;    // [0, 4096)
    const int p0   = tile << 4;
    const int m    = lane & 15;              // row within tile (A) / column (B,D)
    const int hi   = lane >> 4;              // 0: K+0/1,  1: K+2/3
    const int p    = p0 + m;
    const int w    = p & 127;
    const int h    = (p >> 7) & 127;
    const int b    = p >> 14;
    const int w0   = p0 & 127;

    const float* xb   = x   + b * (C_ * HW_);
    const float* offb = off + b * (18 * HW_);

    v8f acc[4];
    acc[0] = v8f{}; acc[1] = v8f{}; acc[2] = v8f{}; acc[3] = v8f{};

    for (int kk = 0; kk < 9; ++kk) {
        const int ki = kk / 3, kj = kk % 3;
        // per-pixel learned offsets
        const float dy = offb[((2 * kk    ) * H_ + h) * W_ + w];
        const float dx = offb[((2 * kk + 1) * H_ + h) * W_ + w];
        const float py = (float)(h - 1 + ki) + dy;
        const float px = (float)(w - 1 + kj) + dx;
        const float y0f = floorf(py), x0f = floorf(px);
        const float ly = py - y0f,   lx = px - x0f;
        const int y0 = (int)y0f, x0 = (int)x0f;
        const int y1 = y0 + 1,   x1 = x0 + 1;
        const float vy0 = ((unsigned)y0 < (unsigned)H_) ? 1.f : 0.f;
        const float vy1 = ((unsigned)y1 < (unsigned)H_) ? 1.f : 0.f;
        const float vx0 = ((unsigned)x0 < (unsigned)W_) ? 1.f : 0.f;
        const float vx1 = ((unsigned)x1 < (unsigned)W_) ? 1.f : 0.f;
        const float w00 = (1.f - ly) * (1.f - lx) * vy0 * vx0;
        const float w01 = (1.f - ly) * lx         * vy0 * vx1;
        const float w10 = ly * (1.f - lx)         * vy1 * vx0;
        const float w11 = ly * lx                 * vy1 * vx1;
        const int yc0 = min(max(y0, 0), H_ - 1), yc1 = min(max(y1, 0), H_ - 1);
        const int xc0 = min(max(x0, 0), W_ - 1), xc1 = min(max(x1, 0), W_ - 1);
        const int t00 = yc0 * W_ + xc0, t01 = yc0 * W_ + xc1;
        const int t10 = yc1 * W_ + xc0, t11 = yc1 * W_ + xc1;

        for (int cc = 0; cc < 16; ++cc) {
            const int c0  = cc << 2;
            const int clo = c0 + (hi << 1);          // this lane's first channel
            const float* xa  = xb + clo * HW_;
            const float* xc2 = xa + HW_;
            v2f a;
            a.x = w00 * xa[t00]  + w01 * xa[t01]  + w10 * xa[t10]  + w11 * xa[t11];
            a.y = w00 * xc2[t00] + w01 * xc2[t01] + w10 * xc2[t10] + w11 * xc2[t11];

            // K-pair index of this lane's B rows; fragment = contiguous b64.
            const int pi = kk * 32 + (c0 >> 1) + hi;
            const float* wr = &wlds[pi * 128 + m * 2];
            #pragma unroll
            for (int nt = 0; nt < 4; ++nt) {
                v2f bf = *(const v2f*)(wr + nt * 32);
                acc[nt] = __builtin_amdgcn_wmma_f32_16x16x4_f32(
                    false, a, false, bf, (short)0, acc[nt], false, false);
            }
        }
    }

    // D layout: lane<16 -> rows v, lane>=16 -> rows v+8; column = nt*16 + m.
    // Rows = consecutive w positions -> two 16B-contiguous stores per tile.
    #pragma unroll
    for (int nt = 0; nt < 4; ++nt) {
        const int o = nt * 16 + m;
        const float bias = b_dcn[o];
        float* op = out + ((b * O_ + o) * H_ + h) * W_ + w0 + (hi << 3);
        float4 s0 = make_float4(acc[nt][0] + bias, acc[nt][1] + bias,
                                acc[nt][2] + bias, acc[nt][3] + bias);
        float4 s1 = make_float4(acc[nt][4] + bias, acc[nt][5] + bias,
                                acc[nt][6] + bias, acc[nt][7] + bias);
        *(float4*)(op)     = s0;
        *(float4*)(op + 4) = s1;
    }
}

// ---------------------------------------------------------------------------
extern "C" void kernel_launch(void* const* d_in, const int* in_sizes, int n_in,
                              void* d_out, int out_size, void* d_ws, size_t ws_size,
                              hipStream_t stream) {
    const float* x     = (const float*)d_in[0];
    const float* w_off = (const float*)d_in[1];
    const float* b_off = (const float*)d_in[2];
    const float* w_dcn = (const float*)d_in[3];
    const float* b_dcn = (const float*)d_in[4];
    float* out = (float*)d_out;

    float* ws_off = (float*)d_ws;              // OFF_ELEMS floats
    float* ws_wt  = ws_off + OFF_ELEMS;        // WT_ELEMS floats

    offset_conv_kernel<<<OFF_ELEMS / 256, 256, 0, stream>>>(x, w_off, b_off, ws_off);
    repack_w_kernel<<<WT_ELEMS / 256, 256, 0, stream>>>(w_dcn, ws_wt);
    deform_wmma_kernel<<<512, 256, 0, stream>>>(x, ws_off, ws_wt, b_dcn, out);
}